// Pct_Seg_1717986918888
// MI455X (gfx1250) — compile-verified
//
#include <hip/hip_runtime.h>

#define NPTS  1024
#define KNN   32
#define BATCH 4
#define EPSBN 1e-5f

typedef __attribute__((ext_vector_type(2))) float v2f;
typedef __attribute__((ext_vector_type(8))) float v8f;

__device__ __forceinline__ v8f wmma4(v2f a, v2f b, v8f c) {
  // D = A(16x4, f32) x B(4x16, f32) + C(16x16, f32)
  return __builtin_amdgcn_wmma_f32_16x16x4_f32(false, a, false, b, (short)0, c, false, false);
}

// ---------------------------------------------------------------------------
// Generic GEMM: Out[b,o,n] = epi( sum_c A[(b),o,c] * X[b,c,n] )
// A batch stride aStride (0 => shared weights), leading dim lda.
// Epilogue: +bias[o], BN(o), /(1e-9+colDiv[b,n]), act(0/1 relu/2 leaky0.2), +residual[b,o,n]
// Block = 256 threads (8 waves) -> 32(M) x 64(N) tile. Grid: (N/64, ceil(O/32), B)
// GUARD=false requires O%32==0 && C%4==0 (branch-free inner loop).
// ---------------------------------------------------------------------------
template <bool GUARD>
__global__ void gemm_wmma_kernel(
    const float* __restrict__ A, size_t aStride, int lda,
    const float* __restrict__ X, size_t xStride,
    float* __restrict__ Out, size_t oStride,
    int O, int C, int N,
    const float* __restrict__ bias,
    const float* __restrict__ bng, const float* __restrict__ bnb,
    const float* __restrict__ bnm, const float* __restrict__ bnv,
    const float* __restrict__ residual,
    const float* __restrict__ colDiv,
    int act)
{
  const int b = blockIdx.z;
  const float* Ab = A + (size_t)b * aStride;
  const float* Xb = X + (size_t)b * xStride;
  float* Ob = Out + (size_t)b * oStride;

  const int tid  = threadIdx.x;
  const int wave = tid >> 5;
  const int lane = tid & 31;
  const int half = lane >> 4;
  const int l16  = lane & 15;

  const int mBase = blockIdx.y * 32 + (wave >> 2) * 16;
  const int nBase = blockIdx.x * 64 + (wave & 3) * 16;
  const int col   = nBase + l16;
  const int rowA  = mBase + l16;

  v8f acc = {0.f,0.f,0.f,0.f,0.f,0.f,0.f,0.f};

  if (!GUARD) {
    // Branch-free hot loop: A as aligned b64, B as two coalesced b32 loads.
    const float* aPtr = Ab + (size_t)rowA * lda + 2 * half;
    const float* bPtr = Xb + (size_t)(2 * half) * N + col;
    for (int k = 0; k < C; k += 4) {
      const v2f a = *(const v2f*)aPtr;
      v2f bb;
      bb.x = bPtr[0];
      bb.y = bPtr[(size_t)N];
      aPtr += 4;
      bPtr += (size_t)4 * N;
      acc = wmma4(a, bb, acc);
    }
  } else {
    // Clamped indices -> loads always in-bounds & unconditional; masks become
    // v_cndmask selects, not EXEC branches.
    const bool rowAok = (rowA < O);
    const int rA = rowAok ? rowA : (O - 1);
    for (int k = 0; k < C; k += 4) {
      const int k0  = k + half * 2;
      const int kc0 = (k0     < C) ? k0     : (C - 1);
      const int kc1 = (k0 + 1 < C) ? k0 + 1 : (C - 1);
      const float a0 = Ab[(size_t)rA * lda + kc0];
      const float a1 = Ab[(size_t)rA * lda + kc1];
      const float b0 = Xb[(size_t)kc0 * N + col];
      const float b1 = Xb[(size_t)kc1 * N + col];
      v2f a, bb;
      a.x  = (rowAok && k0     < C) ? a0 : 0.f;
      a.y  = (rowAok && k0 + 1 < C) ? a1 : 0.f;
      bb.x = (k0     < C) ? b0 : 0.f;
      bb.y = (k0 + 1 < C) ? b1 : 0.f;
      acc = wmma4(a, bb, acc);
    }
  }

  #pragma unroll
  for (int v = 0; v < 8; ++v) {
    const int row = mBase + v + half * 8;
    if (GUARD && row >= O) continue;
    float val = acc[v];
    if (bias) val += bias[row];
    if (bng) {
      const float sc = bng[row] * rsqrtf(bnv[row] + EPSBN);
      val = (val - bnm[row]) * sc + bnb[row];
    }
    if (colDiv) val /= (1e-9f + colDiv[(size_t)b * N + col]);
    if (act == 1)      val = fmaxf(val, 0.f);
    else if (act == 2) val = (val > 0.f) ? val : 0.2f * val;
    if (residual) val += residual[(size_t)b * oStride + (size_t)row * N + col];
    Ob[(size_t)row * N + col] = val;
  }
}

// ---------------------------------------------------------------------------
// Gram: Out[b,n,m] = scale * sum_c X[b,c,n]*X[b,c,m]  (- xx[b,n] - xx[b,m])
// kNN pairwise distance (scale=2, xx) and attention energy (scale=1, xx=null).
// Grid: (N/64, N/32, B), block 256. Requires C%4==0 (always true here).
// ---------------------------------------------------------------------------
__global__ void gram_wmma_kernel(
    const float* __restrict__ X, size_t xStride, int C, int N,
    float* __restrict__ Out, size_t oStride, float scale,
    const float* __restrict__ xx)
{
  const int b = blockIdx.z;
  const float* Xb = X + (size_t)b * xStride;
  float* Ob = Out + (size_t)b * oStride;

  const int tid  = threadIdx.x;
  const int wave = tid >> 5;
  const int lane = tid & 31;
  const int half = lane >> 4;
  const int l16  = lane & 15;

  const int mBase = blockIdx.y * 32 + (wave >> 2) * 16;
  const int nBase = blockIdx.x * 64 + (wave & 3) * 16;

  const float* aPtr = Xb + (size_t)(2 * half) * N + mBase + l16;
  const float* bPtr = Xb + (size_t)(2 * half) * N + nBase + l16;

  v8f acc = {0.f,0.f,0.f,0.f,0.f,0.f,0.f,0.f};
  for (int k = 0; k < C; k += 4) {
    v2f a, bb;
    a.x  = aPtr[0];
    a.y  = aPtr[(size_t)N];
    bb.x = bPtr[0];
    bb.y = bPtr[(size_t)N];
    aPtr += (size_t)4 * N;
    bPtr += (size_t)4 * N;
    acc = wmma4(a, bb, acc);
  }

  const int col = nBase + l16;
  #pragma unroll
  for (int v = 0; v < 8; ++v) {
    const int row = mBase + v + half * 8;
    float val = scale * acc[v];
    if (xx) val -= xx[(size_t)b * N + row] + xx[(size_t)b * N + col];
    Ob[(size_t)row * N + col] = val;
  }
}

// ---------------------------------------------------------------------------
// Edge conv fused: out[b,o,n] = max_k leaky0.2(BN(sum_c W[o,c]*g[b,c,n,k]))
// g[c] = h[c, idx[n,k]] - h[c,n]  (c < C0),  h[c-C0, n]  (c >= C0)
// Two-phase K loop (uniform split at C0) -> no divergent conditionals.
// Grid: (N, O/128, B), block 256 (8 waves, one 16-row o-tile each).
// ---------------------------------------------------------------------------
__global__ void edgeconv_wmma_kernel(
    const float* __restrict__ W,
    const float* __restrict__ H, size_t hStride,
    const int* __restrict__ knnIdx,
    float* __restrict__ Out, size_t oStride,
    int O, int C0,
    const float* __restrict__ bng, const float* __restrict__ bnb,
    const float* __restrict__ bnm, const float* __restrict__ bnv)
{
  const int b = blockIdx.z;
  const int n = blockIdx.x;
  const float* Hb = H + (size_t)b * hStride;

  const int tid  = threadIdx.x;
  const int wave = tid >> 5;
  const int lane = tid & 31;
  const int half = lane >> 4;
  const int l16  = lane & 15;
  const int oTile = blockIdx.y * 8 + wave;

  __shared__ float scen[128];
  __shared__ int   sj[KNN];
  if (tid < C0)  scen[tid] = Hb[(size_t)tid * NPTS + n];
  if (tid < KNN) sj[tid]   = knnIdx[((size_t)b * NPTS + n) * KNN + tid];
  __syncthreads();

  const int jL = sj[l16];
  const int jH = sj[l16 + 16];
  const int C2 = 2 * C0;
  const int rowA = oTile * 16 + l16;
  const int k0base = 2 * half;

  const float* aPtr = W + (size_t)rowA * C2 + k0base;
  const float* hLp  = Hb + (size_t)k0base * NPTS + jL;
  const float* hHp  = Hb + (size_t)k0base * NPTS + jH;

  v8f accL = {0.f,0.f,0.f,0.f,0.f,0.f,0.f,0.f};
  v8f accH = {0.f,0.f,0.f,0.f,0.f,0.f,0.f,0.f};

  // Phase 1: feature-minus-center channels (gathers).
  for (int k = 0; k < C0; k += 4) {
    const int k0 = k + k0base;
    const v2f a = *(const v2f*)aPtr;
    const float c0 = scen[k0];
    const float c1 = scen[k0 + 1];
    v2f bL, bH;
    bL.x = hLp[0]            - c0;
    bL.y = hLp[(size_t)NPTS] - c1;
    bH.x = hHp[0]            - c0;
    bH.y = hHp[(size_t)NPTS] - c1;
    aPtr += 4;
    hLp  += (size_t)4 * NPTS;
    hHp  += (size_t)4 * NPTS;
    accL = wmma4(a, bL, accL);
    accH = wmma4(a, bH, accH);
  }
  // Phase 2: broadcast-center channels (LDS only, no global loads).
  for (int k = 0; k < C0; k += 4) {
    const int k0 = k + k0base;
    const v2f a = *(const v2f*)aPtr;
    aPtr += 4;
    v2f bc;
    bc.x = scen[k0];
    bc.y = scen[k0 + 1];
    accL = wmma4(a, bc, accL);
    accH = wmma4(a, bc, accH);
  }

  float* Ob = Out + (size_t)b * oStride;
  #pragma unroll
  for (int v = 0; v < 8; ++v) {
    const int row = oTile * 16 + v + half * 8;
    const float sc = bng[row] * rsqrtf(bnv[row] + EPSBN);
    float xL = (accL[v] - bnm[row]) * sc + bnb[row];
    float xH = (accH[v] - bnm[row]) * sc + bnb[row];
    xL = (xL > 0.f) ? xL : 0.2f * xL;
    xH = (xH > 0.f) ? xH : 0.2f * xH;
    float m = fmaxf(xL, xH);
    #pragma unroll
    for (int off = 8; off > 0; off >>= 1)
      m = fmaxf(m, __shfl_xor(m, off, 16));
    if (l16 == 0) Ob[(size_t)row * NPTS + n] = m;
  }
}

// ---------------------------------------------------------------------------
// xx[b,n] = sum_c X[b,c,n]^2
// ---------------------------------------------------------------------------
__global__ void sumsq_kernel(const float* __restrict__ X, size_t xStride, int C,
                             float* __restrict__ xx)
{
  const int b = blockIdx.y;
  const int n = blockIdx.x * 256 + threadIdx.x;
  const float* Xb = X + (size_t)b * xStride;
  float s = 0.f;
  for (int c = 0; c < C; ++c) { float v = Xb[(size_t)c * NPTS + n]; s += v * v; }
  xx[(size_t)b * NPTS + n] = s;
}

// ---------------------------------------------------------------------------
// Top-32 per row (largest values; tie -> smallest index, like lax.top_k).
// Grid: (N, B), block 256.
// ---------------------------------------------------------------------------
__global__ void topk_kernel(const float* __restrict__ pd, int* __restrict__ idxOut)
{
  const int b = blockIdx.y, n = blockIdx.x, tid = threadIdx.x;
  const float* row = pd + ((size_t)b * NPTS + n) * NPTS;
  __shared__ float sval[NPTS];
  __shared__ float rv[256];
  __shared__ int   ri[256];
  for (int i = tid; i < NPTS; i += 256) sval[i] = row[i];
  __syncthreads();
  for (int t = 0; t < KNN; ++t) {
    float bv = -3.4e38f; int bi = 0;
    const int base = tid * 4;
    #pragma unroll
    for (int i = 0; i < 4; ++i) {
      float v = sval[base + i];
      if (v > bv) { bv = v; bi = base + i; }
    }
    rv[tid] = bv; ri[tid] = bi;
    __syncthreads();
    for (int s = 128; s > 0; s >>= 1) {
      if (tid < s) {
        if (rv[tid+s] > rv[tid] || (rv[tid+s] == rv[tid] && ri[tid+s] < ri[tid])) {
          rv[tid] = rv[tid+s]; ri[tid] = ri[tid+s];
        }
      }
      __syncthreads();
    }
    if (tid == 0) { idxOut[((size_t)b * NPTS + n) * KNN + t] = ri[0]; sval[ri[0]] = -3.4e38f; }
    __syncthreads();
  }
}

// ---------------------------------------------------------------------------
// Row softmax (in place) over last axis. Grid: (N, B), block 256.
// ---------------------------------------------------------------------------
__global__ void softmax_rows_kernel(float* __restrict__ E)
{
  const int b = blockIdx.y, n = blockIdx.x, tid = threadIdx.x;
  float* row = E + ((size_t)b * NPTS + n) * NPTS;
  __shared__ float red[256];
  __shared__ float bc;
  float m = -3.4e38f;
  for (int i = tid; i < NPTS; i += 256) m = fmaxf(m, row[i]);
  red[tid] = m; __syncthreads();
  for (int s = 128; s > 0; s >>= 1) { if (tid < s) red[tid] = fmaxf(red[tid], red[tid+s]); __syncthreads(); }
  if (tid == 0) bc = red[0];
  __syncthreads();
  m = bc;
  __syncthreads();
  float sum = 0.f;
  for (int i = tid; i < NPTS; i += 256) { float e = __expf(row[i] - m); row[i] = e; sum += e; }
  red[tid] = sum; __syncthreads();
  for (int s = 128; s > 0; s >>= 1) { if (tid < s) red[tid] += red[tid+s]; __syncthreads(); }
  if (tid == 0) bc = red[0];
  __syncthreads();
  const float inv = 1.f / bc;
  for (int i = tid; i < NPTS; i += 256) row[i] *= inv;
}

// cs[b,m] = sum_n A[b,n,m]
__global__ void colsum_kernel(const float* __restrict__ A, float* __restrict__ cs)
{
  const int b = blockIdx.y;
  const int m = blockIdx.x * 256 + threadIdx.x;
  const float* Ab = A + (size_t)b * NPTS * NPTS;
  float s = 0.f;
  for (int n = 0; n < NPTS; ++n) s += Ab[(size_t)n * NPTS + m];
  cs[(size_t)b * NPTS + m] = s;
}

// out[b,c] = max_n H[b,c,n].  Grid: (C, B), block 256.
__global__ void rowmax_kernel(const float* __restrict__ H, size_t hStride, int C,
                              float* __restrict__ out)
{
  const int b = blockIdx.y, c = blockIdx.x, tid = threadIdx.x;
  const float* row = H + (size_t)b * hStride + (size_t)c * NPTS;
  __shared__ float red[256];
  float m = -3.4e38f;
  for (int i = tid; i < NPTS; i += 256) m = fmaxf(m, row[i]);
  red[tid] = m; __syncthreads();
  for (int s = 128; s > 0; s >>= 1) { if (tid < s) red[tid] = fmaxf(red[tid], red[tid+s]); __syncthreads(); }
  if (tid == 0) out[(size_t)b * C + c] = red[0];
}

__global__ void sub_kernel(const float* __restrict__ X, const float* __restrict__ Y,
                           float* __restrict__ Z, size_t total)
{
  size_t i = (size_t)blockIdx.x * 256 + threadIdx.x;
  if (i < total) Z[i] = X[i] - Y[i];
}

// cat over channel: [h1,h2,h3,h4,f1] -> (B,1280,N)
__global__ void concat5_kernel(const float* __restrict__ h1, const float* __restrict__ h2,
                               const float* __restrict__ h3, const float* __restrict__ h4,
                               const float* __restrict__ f1, float* __restrict__ out)
{
  size_t i = (size_t)blockIdx.x * 256 + threadIdx.x;
  const size_t total = (size_t)BATCH * 1280 * NPTS;
  if (i >= total) return;
  const int n  = (int)(i % NPTS);
  const size_t t = i / NPTS;
  const int ch = (int)(t % 1280);
  const int b  = (int)(t / 1280);
  const int g  = ch >> 8;
  const int c  = ch & 255;
  const float* src = (g == 0) ? h1 : (g == 1) ? h2 : (g == 2) ? h3 : (g == 3) ? h4 : f1;
  out[i] = src[((size_t)b * 256 + c) * NPTS + n];
}

// cat: [broadcast(gmax (B,1024)), x2 (B,256,N)] -> (B,1280,N)
__global__ void concat_gmax_kernel(const float* __restrict__ gmax, const float* __restrict__ x2,
                                   float* __restrict__ out)
{
  size_t i = (size_t)blockIdx.x * 256 + threadIdx.x;
  const size_t total = (size_t)BATCH * 1280 * NPTS;
  if (i >= total) return;
  const int n  = (int)(i % NPTS);
  const size_t t = i / NPTS;
  const int ch = (int)(t % 1280);
  const int b  = (int)(t / 1280);
  out[i] = (ch < 1024) ? gmax[(size_t)b * 1024 + ch]
                       : x2[((size_t)b * 256 + (ch - 1024)) * NPTS + n];
}

// ---------------------------------------------------------------------------
// Host side
// ---------------------------------------------------------------------------
namespace {

struct BNP { const float *g, *b, *m, *v; };

inline void gemm(hipStream_t s, const float* A, size_t aStride, int lda,
                 const float* X, size_t xStride, float* Out, size_t oStride,
                 int O, int C,
                 const float* bias, const BNP* bn, const float* residual,
                 const float* colDiv, int act)
{
  dim3 grid(NPTS / 64, (O + 31) / 32, BATCH), block(256);
  const bool guard = (O % 32 != 0) || (C % 4 != 0);
  if (guard) {
    gemm_wmma_kernel<true><<<grid, block, 0, s>>>(
        A, aStride, lda, X, xStride, Out, oStride, O, C, NPTS,
        bias,
        bn ? bn->g : nullptr, bn ? bn->b : nullptr,
        bn ? bn->m : nullptr, bn ? bn->v : nullptr,
        residual, colDiv, act);
  } else {
    gemm_wmma_kernel<false><<<grid, block, 0, s>>>(
        A, aStride, lda, X, xStride, Out, oStride, O, C, NPTS,
        bias,
        bn ? bn->g : nullptr, bn ? bn->b : nullptr,
        bn ? bn->m : nullptr, bn ? bn->v : nullptr,
        residual, colDiv, act);
  }
}

} // namespace

extern "C" void kernel_launch(void* const* d_in, const int* in_sizes, int n_in,
                              void* d_out, int out_size, void* d_ws, size_t ws_size,
                              hipStream_t stream)
{
  (void)in_sizes; (void)n_in; (void)out_size; (void)ws_size;
  auto F = [&](int i) { return (const float*)d_in[i]; };

  // JAX pytree (sorted dict keys) flattening of {'params':{...}, 'x':...}:
  // bn1,bn2,bn201,bn202,bn203,bn3,bn4,bnf,bnp1,bnp2 (4 leaves each),
  // sa1..sa4 (bn.g/b/m/v, bt, bv, wqk, wt, wv),
  // w1,w2,w201,w202,w203,w204,wf,wg1,wg2,wp1,wp2, x
  BNP bn1  {F(0),F(1),F(2),F(3)},     bn2  {F(4),F(5),F(6),F(7)};
  BNP bn201{F(8),F(9),F(10),F(11)},   bn202{F(12),F(13),F(14),F(15)};
  BNP bn203{F(16),F(17),F(18),F(19)}, bn3  {F(20),F(21),F(22),F(23)};
  BNP bn4  {F(24),F(25),F(26),F(27)}, bnf  {F(28),F(29),F(30),F(31)};
  BNP bnp1 {F(32),F(33),F(34),F(35)}, bnp2 {F(36),F(37),F(38),F(39)};

  struct SAP { BNP bn; const float *bt, *bv, *wqk, *wt, *wv; } sa[4];
  for (int i = 0; i < 4; ++i) {
    const int base = 40 + 9 * i;
    sa[i].bn  = BNP{F(base), F(base+1), F(base+2), F(base+3)};
    sa[i].bt  = F(base+4); sa[i].bv = F(base+5);
    sa[i].wqk = F(base+6); sa[i].wt = F(base+7); sa[i].wv = F(base+8);
  }
  const float *w1 = F(76), *w2 = F(77), *w201 = F(78), *w202 = F(79), *w203 = F(80), *w204 = F(81);
  const float *wf = F(82), *wg1 = F(83), *wg2 = F(84), *wp1 = F(85), *wp2 = F(86), *x = F(87);

  // Workspace bump allocator (256B aligned).
  char* wsp = (char*)d_ws;
  auto alloc = [&](size_t elems) -> float* {
    float* r = (float*)wsp;
    wsp += ((elems * sizeof(float)) + 255) & ~(size_t)255;
    return r;
  };
  float* pd   = alloc((size_t)BATCH * NPTS * NPTS);      // pairwise / energy / attn
  int*   kidx = (int*)alloc((size_t)BATCH * NPTS * KNN);
  float* xx   = alloc((size_t)BATCH * NPTS);
  float* t0   = alloc((size_t)BATCH * 64 * NPTS);
  float* t1   = alloc((size_t)BATCH * 64 * NPTS);
  float* x1   = alloc((size_t)BATCH * 128 * NPTS);
  float* f1   = alloc((size_t)BATCH * 256 * NPTS);
  float* p1   = alloc((size_t)BATCH * 256 * NPTS);
  float* p2   = alloc((size_t)BATCH * 256 * NPTS);
  float* qk   = alloc((size_t)BATCH * 64 * NPTS);
  float* xv   = alloc((size_t)BATCH * 256 * NPTS);
  float* cs   = alloc((size_t)BATCH * NPTS);
  float* xr   = alloc((size_t)BATCH * 256 * NPTS);
  float* dif  = alloc((size_t)BATCH * 256 * NPTS);
  float* hsa[4];
  for (int i = 0; i < 4; ++i) hsa[i] = alloc((size_t)BATCH * 256 * NPTS);
  float* cat1 = alloc((size_t)BATCH * 1280 * NPTS);
  float* hf   = alloc((size_t)BATCH * 1024 * NPTS);
  float* x2b  = alloc((size_t)BATCH * 256 * NPTS);
  float* gmx  = alloc((size_t)BATCH * 1024);
  float* cat2 = alloc((size_t)BATCH * 1280 * NPTS);
  float* h202 = alloc((size_t)BATCH * 256 * NPTS);
  float* h203 = alloc((size_t)BATCH * 256 * NPTS);

  const dim3 blk(256);
  const size_t NN = (size_t)NPTS * NPTS;

  // Stage 1: mlp1 + mlp2
  gemm(stream, w1, 0, 3,  x,  (size_t)3 * NPTS,  t0, (size_t)64 * NPTS, 64, 3,  nullptr, &bn1, nullptr, nullptr, 1);
  gemm(stream, w2, 0, 64, t0, (size_t)64 * NPTS, t1, (size_t)64 * NPTS, 64, 64, nullptr, &bn2, nullptr, nullptr, 1);

  // Edge conv 1 (C0=64 -> O=128)
  sumsq_kernel<<<dim3(NPTS/256, BATCH), blk, 0, stream>>>(t1, (size_t)64 * NPTS, 64, xx);
  gram_wmma_kernel<<<dim3(NPTS/64, NPTS/32, BATCH), blk, 0, stream>>>(
      t1, (size_t)64 * NPTS, 64, NPTS, pd, NN, 2.f, xx);
  topk_kernel<<<dim3(NPTS, BATCH), blk, 0, stream>>>(pd, kidx);
  edgeconv_wmma_kernel<<<dim3(NPTS, 1, BATCH), blk, 0, stream>>>(
      wg1, t1, (size_t)64 * NPTS, kidx, x1, (size_t)128 * NPTS, 128, 64,
      bn3.g, bn3.b, bn3.m, bn3.v);

  // Edge conv 2 (C0=128 -> O=256)
  sumsq_kernel<<<dim3(NPTS/256, BATCH), blk, 0, stream>>>(x1, (size_t)128 * NPTS, 128, xx);
  gram_wmma_kernel<<<dim3(NPTS/64, NPTS/32, BATCH), blk, 0, stream>>>(
      x1, (size_t)128 * NPTS, 128, NPTS, pd, NN, 2.f, xx);
  topk_kernel<<<dim3(NPTS, BATCH), blk, 0, stream>>>(pd, kidx);
  edgeconv_wmma_kernel<<<dim3(NPTS, 2, BATCH), blk, 0, stream>>>(
      wg2, x1, (size_t)128 * NPTS, kidx, f1, (size_t)256 * NPTS, 256, 128,
      bn4.g, bn4.b, bn4.m, bn4.v);

  // Pre-attention MLPs
  gemm(stream, wp1, 0, 256, f1, (size_t)256 * NPTS, p1, (size_t)256 * NPTS, 256, 256, nullptr, &bnp1, nullptr, nullptr, 1);
  gemm(stream, wp2, 0, 256, p1, (size_t)256 * NPTS, p2, (size_t)256 * NPTS, 256, 256, nullptr, &bnp2, nullptr, nullptr, 1);

  // Self-attention stack
  const float* hin = p2;
  const size_t eTot = (size_t)BATCH * 256 * NPTS;
  for (int i = 0; i < 4; ++i) {
    gemm(stream, sa[i].wqk, 0, 256, hin, (size_t)256 * NPTS, qk, (size_t)64 * NPTS, 64, 256,
         nullptr, nullptr, nullptr, nullptr, 0);
    gemm(stream, sa[i].wv, 0, 256, hin, (size_t)256 * NPTS, xv, (size_t)256 * NPTS, 256, 256,
         sa[i].bv, nullptr, nullptr, nullptr, 0);
    // energy = qk^T qk (shared q/k projection -> symmetric)
    gram_wmma_kernel<<<dim3(NPTS/64, NPTS/32, BATCH), blk, 0, stream>>>(
        qk, (size_t)64 * NPTS, 64, NPTS, pd, NN, 1.f, nullptr);
    softmax_rows_kernel<<<dim3(NPTS, BATCH), blk, 0, stream>>>(pd);
    colsum_kernel<<<dim3(NPTS/256, BATCH), blk, 0, stream>>>(pd, cs);
    // x_r = (xv @ attn) / (1e-9 + colsum[m])   (batch-strided A)
    gemm(stream, xv, (size_t)256 * NPTS, NPTS, pd, NN, xr, (size_t)256 * NPTS, 256, NPTS,
         nullptr, nullptr, nullptr, cs, 0);
    sub_kernel<<<(unsigned)((eTot + 255) / 256), blk, 0, stream>>>(hin, xr, dif, eTot);
    // hout = hin + relu(BN(wt @ (hin - xr) + bt))
    gemm(stream, sa[i].wt, 0, 256, dif, (size_t)256 * NPTS, hsa[i], (size_t)256 * NPTS, 256, 256,
         sa[i].bt, &sa[i].bn, hin, nullptr, 1);
    hin = hsa[i];
  }

  // Fuse + head
  const size_t cTot = (size_t)BATCH * 1280 * NPTS;
  concat5_kernel<<<(unsigned)((cTot + 255) / 256), blk, 0, stream>>>(hsa[0], hsa[1], hsa[2], hsa[3], f1, cat1);
  gemm(stream, wf,   0, 1280, cat1, (size_t)1280 * NPTS, hf,  (size_t)1024 * NPTS, 1024, 1280, nullptr, &bnf,   nullptr, nullptr, 2);
  gemm(stream, w201, 0, 1024, hf,   (size_t)1024 * NPTS, x2b, (size_t)256  * NPTS, 256,  1024, nullptr, &bn201, nullptr, nullptr, 2);
  rowmax_kernel<<<dim3(1024, BATCH), blk, 0, stream>>>(hf, (size_t)1024 * NPTS, 1024, gmx);
  concat_gmax_kernel<<<(unsigned)((cTot + 255) / 256), blk, 0, stream>>>(gmx, x2b, cat2);
  gemm(stream, w202, 0, 1280, cat2, (size_t)1280 * NPTS, h202, (size_t)256 * NPTS, 256, 1280, nullptr, &bn202, nullptr, nullptr, 2);
  gemm(stream, w203, 0, 256,  h202, (size_t)256  * NPTS, h203, (size_t)256 * NPTS, 256, 256,  nullptr, &bn203, nullptr, nullptr, 2);
  gemm(stream, w204, 0, 256,  h203, (size_t)256  * NPTS, (float*)d_out, (size_t)50 * NPTS, 50, 256,
       nullptr, nullptr, nullptr, nullptr, 0);
}